// BlockDiagonalProjection_8521215115565
// MI455X (gfx1250) — compile-verified
//
#include <hip/hip_runtime.h>
#include <hip/hip_bf16.h>

// Block-diagonal projection: out[b,s,h,o] = sum_i in[b,s,h,i] * W[h,o,i]
// B=8, S=2048, H=16, N_IN=N_OUT=128, fp32 in/out.
// fp32 WMMA pipe (V_WMMA_F32_16X16X4_F32); W[h] staged in LDS K-pair-
// interleaved so each lane's B fragment is a single conflict-free ds_load_b64.

typedef __attribute__((ext_vector_type(2))) float v2f;
typedef __attribute__((ext_vector_type(8))) float v8f;

#define BDP_H     16
#define BDP_NIN   128
#define BDP_NOUT  128
// float2 row pitch; 2*PITCH dwords = 288 ≡ 32 (mod 64 banks):
// lanes 0-15 (row p) hit banks 0..31, lanes 16-31 (row p+1) hit banks 32..63.
#define BT_PITCH  144
#define TOK_PER_WG 128  // 8 waves * 16 tokens

__global__ __launch_bounds__(256)
void bdp_wmma_f32_kernel(const float* __restrict__ in,
                         const float* __restrict__ W,
                         float* __restrict__ out) {
    // BT[p][o] = (W[h][o][2p], W[h][o][2p+1]) : B-fragment for K={2p,2p+1}, col o
    __shared__ v2f BT[64 * BT_PITCH];   // 72 KB

    const int h        = (int)blockIdx.x & (BDP_H - 1);
    const int tokBlock = (int)blockIdx.x >> 4;
    const int tid      = (int)threadIdx.x;

    // ---- Fill: 8192 float2 pairs. Consecutive lanes take consecutive o at
    // fixed p -> LDS store stride 8B (conflict-free). Global reads are strided
    // but W[h] is 64KB and L2-resident (shared by 128 WGs per head).
    const v2f* Wh2 = (const v2f*)(W + h * (BDP_NOUT * BDP_NIN));
    #pragma unroll
    for (int it = 0; it < 32; ++it) {
        const int idx = tid + it * 256;
        const int o = idx & 127;        // output column
        const int p = idx >> 7;         // K-pair index
        BT[p * BT_PITCH + o] = Wh2[o * 64 + p];
    }
    __syncthreads();

    const int wave = tid >> 5;
    const int lane = tid & 31;
    const int nrow = lane & 15;         // M for A, N for B/C
    const int kh   = lane >> 4;         // 0: K={0,1}; 1: K={2,3} within each quad

    const int tokBase = tokBlock * TOK_PER_WG + wave * 16;

    // ---- A tile: 16 tokens x 128 K, loaded once (32 v2f fragments) ----
    // fp32 A layout (16x4): lanes 0-15 hold K=0,1; lanes 16-31 hold K=2,3.
    const float* Arow = in + ((tokBase + nrow) * BDP_H + h) * BDP_NIN + kh * 2;
    v2f a[32];
    #pragma unroll
    for (int k = 0; k < 32; ++k) {
        a[k] = *(const v2f*)(Arow + 4 * k);
    }

    // ---- 8 N-tiles x 32 K-step WMMA chain ----
    #pragma unroll
    for (int nt = 0; nt < 8; ++nt) {
        v8f c = {};
        // lane's fragment: row p = 2k + kh, column nt*16 + nrow -> one b64
        const v2f* Bb = &BT[kh * BT_PITCH + nt * 16 + nrow];
        #pragma unroll
        for (int k = 0; k < 32; ++k) {
            const v2f b = Bb[(2 * k) * BT_PITCH];
            // (neg_a, A, neg_b, B, c_mod, C, reuse_a, reuse_b)
            c = __builtin_amdgcn_wmma_f32_16x16x4_f32(
                    false, a[k], false, b, (short)0, c, false, false);
        }
        // C/D layout: VGPR r -> M = r + (lane>=16 ? 8 : 0), N = nt*16 + (lane&15)
        const int n  = nt * 16 + nrow;
        const int mb = kh * 8;
        #pragma unroll
        for (int r = 0; r < 8; ++r) {
            out[((tokBase + mb + r) * BDP_H + h) * BDP_NOUT + n] = c[r];
        }
    }
}

extern "C" void kernel_launch(void* const* d_in, const int* in_sizes, int n_in,
                              void* d_out, int out_size, void* d_ws, size_t ws_size,
                              hipStream_t stream) {
    const float* in  = (const float*)d_in[0];   // [8,2048,16,128] f32
    const float* W   = (const float*)d_in[1];   // [16,128,128]   f32
    float*       out = (float*)d_out;           // [8,2048,16,128] f32
    (void)in_sizes; (void)n_in; (void)out_size; (void)d_ws; (void)ws_size;

    const int tokTotal  = 8 * 2048;                        // 16384 tokens
    const int numBlocks = (tokTotal / TOK_PER_WG) * BDP_H; // 128 * 16 = 2048
    dim3 grid(numBlocks), block(256);
    bdp_wmma_f32_kernel<<<grid, block, 0, stream>>>(in, W, out);
}